// TATT_1_3556232921192
// MI455X (gfx1250) — compile-verified
//
#include <hip/hip_runtime.h>
#include <math.h>

constexpr int B = 8, C = 512, N = 2048, T = 24;
constexpr int NN  = 128;       // n-tile width in K1
constexpr int NCH = N / NN;    // 16 n-tiles
constexpr int CS  = 4;         // c splits (parallelism for K1)
constexpr int CPS = C / CS;    // 128 c's per block

typedef float v2f __attribute__((ext_vector_type(2)));
typedef float v8f __attribute__((ext_vector_type(8)));

// --------------------------------------------------------------------------
// K1: single streaming pass over seq (805 MB). Async global->LDS staging
// (double buffered on ASYNCcnt), fused dual GEMV:
//   f1part[s,b,t,n] = sum_{c in split s} w1[c]*seq[b,c,n,t]
//   f2part[b,ch,c,t] = sum_{n in tile ch} w2[n]*seq[b,c,n,t]
// --------------------------------------------------------------------------
__global__ __launch_bounds__(256) void k1_stream(
    const float* __restrict__ seq, const float* __restrict__ w1,
    const float* __restrict__ w2, float* __restrict__ f1part,
    float* __restrict__ f2part)
{
  __shared__ __align__(16) float buf[2][NN * T];   // 2 x 12 KB
  __shared__ float lw2[NN];
  __shared__ float red[10 * T];

  const int tid   = threadIdx.x;
  const int bx    = blockIdx.x;
  const int split = bx >> 7;        // 0..3
  const int rem   = bx & 127;
  const int b     = rem >> 4;       // 0..7
  const int chunk = rem & 15;       // 0..15
  const int n0    = chunk * NN;
  const int c0    = split * CPS;

  for (int i = tid; i < NN; i += 256) lw2[i] = w2[n0 + i];

  // issue per-thread async b128 copies of column c into LDS buffer bi
  auto issue = [&](int c, int bi) {
    const int base = ((b * C + c) * N + n0) * T;   // floats; < 2^31
#pragma unroll
    for (int k = 0; k < 3; ++k) {
      const int p4 = tid + 256 * k;                // 0..767 float4 slots
      const float* g = seq + base + p4 * 4;
      unsigned loff = (unsigned)(size_t)(&buf[bi][p4 * 4]); // low 32b = LDS addr
      asm volatile("global_load_async_to_lds_b128 %0, %1, off"
                   :: "v"(loff), "v"(g) : "memory");
    }
  };

  issue(c0, 0);
  float4 acc[3] = {};

  for (int cc = 0; cc < CPS; ++cc) {
    const int c  = c0 + cc;
    const int bi = cc & 1;
    if (cc + 1 < CPS) {
      issue(c + 1, bi ^ 1);                       // prefetch next column
      asm volatile("s_wait_asynccnt 3" ::: "memory"); // oldest 3 (this buf) done
    } else {
      asm volatile("s_wait_asynccnt 0" ::: "memory");
    }
    __syncthreads();

    // ---- f1: register accumulation over c, fixed (n,t) slots per thread ----
    const float w1c = w1[c];
#pragma unroll
    for (int k = 0; k < 3; ++k) {
      const float4 v4 = *(const float4*)&buf[bi][(tid + 256 * k) * 4];
      acc[k].x += w1c * v4.x;  acc[k].y += w1c * v4.y;
      acc[k].z += w1c * v4.z;  acc[k].w += w1c * v4.w;
    }

    // ---- f2: reduce over n-tile into 24 t-buckets (LDS tree) ----
    const int t = tid % T;
    const int g = tid / T;     // 0..10 (g<10 active: 10 groups of n)
    if (g < 10) {
      float a2 = 0.f;
      for (int n = g; n < NN; n += 10) a2 += lw2[n] * buf[bi][n * T + t];
      red[g * T + t] = a2;
    }
    __syncthreads();
    if (tid < T) {
      float s = 0.f;
#pragma unroll
      for (int gg = 0; gg < 10; ++gg) s += red[gg * T + tid];
      f2part[((b * NCH + chunk) * C + c) * T + tid] = s;
    }
    __syncthreads();   // protects red rewrite and buf reuse by next issue()
  }

  // write f1 partials (scatter transpose to (split,b,t,n))
#pragma unroll
  for (int k = 0; k < 3; ++k) {
    const int p = (tid + 256 * k) * 4;
    const float vals[4] = {acc[k].x, acc[k].y, acc[k].z, acc[k].w};
#pragma unroll
    for (int j = 0; j < 4; ++j) {
      const int pp = p + j;
      const int n = pp / T, tt = pp % T;
      f1part[((split * B + b) * T + tt) * N + n0 + n] = vals[j];
    }
  }
}

// --------------------------------------------------------------------------
// K2a: f2r[b,c,t] = sum over 16 n-tile partials
// --------------------------------------------------------------------------
__global__ __launch_bounds__(256) void k2a_reduce(
    const float* __restrict__ f2part, float* __restrict__ f2r)
{
  const int idx = blockIdx.x * 256 + threadIdx.x;   // exactly B*C*T = 98304
  if (idx >= B * C * T) return;
  const int b = idx / (C * T);
  const int r = idx % (C * T);
  float s = 0.f;
  for (int ch = 0; ch < NCH; ++ch) s += f2part[(b * NCH + ch) * (C * T) + r];
  f2r[idx] = s;
}

// --------------------------------------------------------------------------
// K2b: dilated conv over time: f2c[b,co,t] = sum_ci W0*f2r[b,ci,t-1]+W1*f2r[b,ci,t+1]
// --------------------------------------------------------------------------
__global__ __launch_bounds__(256) void k2b_convd2(
    const float* __restrict__ f2r, const float* __restrict__ Wd2,
    float* __restrict__ f2c)
{
  __shared__ float fr[C * T];               // 48 KB
  const int b   = blockIdx.x >> 3;
  const int co0 = (blockIdx.x & 7) * 64;
  for (int i = threadIdx.x; i < C * T; i += 256) fr[i] = f2r[b * C * T + i];
  __syncthreads();
#pragma unroll
  for (int k = 0; k < 6; ++k) {
    const int o  = k * 256 + threadIdx.x;   // 1536 outputs per block
    const int co = co0 + o / T;
    const int t  = o % T;
    const float* wrow = Wd2 + co * C * 2;   // Wd2: (C, C, 2)
    float a = 0.f;
    for (int ci = 0; ci < C; ++ci) {
      const float xm = (t > 0)     ? fr[ci * T + t - 1] : 0.f;
      const float xp = (t < T - 1) ? fr[ci * T + t + 1] : 0.f;
      a += wrow[2 * ci] * xm + wrow[2 * ci + 1] * xp;
    }
    f2c[(b * C + co) * T + t] = a;
  }
}

// --------------------------------------------------------------------------
// K3: combine f1 c-split partials + dilated conv over node axis:
//   f1c[b,to,n] = sum_ti Wd1[to,ti,0]*f1[b,ti,n-1] + Wd1[to,ti,1]*f1[b,ti,n+1]
// --------------------------------------------------------------------------
__global__ __launch_bounds__(256) void k3_convd1(
    const float* __restrict__ f1part, const float* __restrict__ Wd1,
    float* __restrict__ f1c)
{
  __shared__ float fs[T * (NN + 2)];        // 24 x 130 halo tile
  __shared__ float wd[T * T * 2];
  const int b     = blockIdx.x >> 4;
  const int chunk = blockIdx.x & 15;
  const int n0    = chunk * NN;
  for (int i = threadIdx.x; i < T * T * 2; i += 256) wd[i] = Wd1[i];
  for (int i = threadIdx.x; i < T * (NN + 2); i += 256) {
    const int ti = i / (NN + 2);
    const int nl = i % (NN + 2);
    const int n  = n0 - 1 + nl;
    float v = 0.f;
    if (n >= 0 && n < N) {
#pragma unroll
      for (int s = 0; s < CS; ++s) v += f1part[((s * B + b) * T + ti) * N + n];
    }
    fs[i] = v;
  }
  __syncthreads();
  for (int o = threadIdx.x; o < T * NN; o += 256) {
    const int to = o / NN;
    const int nl = o % NN;                  // global n = n0+nl ; fs col nl+1
    float a = 0.f;
#pragma unroll
    for (int ti = 0; ti < T; ++ti)
      a += wd[(to * T + ti) * 2 + 0] * fs[ti * (NN + 2) + nl]
         + wd[(to * T + ti) * 2 + 1] * fs[ti * (NN + 2) + nl + 2];
    f1c[(b * T + to) * N + n0 + nl] = a;
  }
}

// --------------------------------------------------------------------------
// K4: X(192x512) = f1c(192x2048) @ w(2048x512) with V_WMMA_F32_16X16X4_F32.
// One wave per 16x16 tile; ISA layouts: A lanes0-15 hold K{0,1}, 16-31 K{2,3};
// C/D vgpr r -> rows r (lanes0-15) / r+8 (lanes16-31).
// --------------------------------------------------------------------------
__global__ __launch_bounds__(32) void k4_wmma_gemm(
    const float* __restrict__ A, const float* __restrict__ Wmat,
    float* __restrict__ X)
{
  const int tile = blockIdx.x;
  const int m0 = (tile >> 5) * 16;          // 12 m-tiles (M=192)
  const int n0 = (tile & 31) * 16;          // 32 n-tiles (Nc=512)
  const int lane = threadIdx.x;
  const int ml = lane & 15;
  const int kh = lane >> 4;
  v8f acc = {0.f, 0.f, 0.f, 0.f, 0.f, 0.f, 0.f, 0.f};
  const float* arow = A + (m0 + ml) * 2048;
  for (int k = 0; k < 2048; k += 4) {
    v2f a, bb;
    a.x  = arow[k + 2 * kh];
    a.y  = arow[k + 2 * kh + 1];
    bb.x = Wmat[(k + 2 * kh) * 512 + n0 + ml];
    bb.y = Wmat[(k + 2 * kh + 1) * 512 + n0 + ml];
    acc = __builtin_amdgcn_wmma_f32_16x16x4_f32(
        /*neg_a=*/false, a, /*neg_b=*/false, bb,
        /*c_mod=*/(short)0, acc, /*reuse_a=*/false, /*reuse_b=*/false);
  }
#pragma unroll
  for (int r = 0; r < 8; ++r)
    X[(m0 + r + 8 * kh) * 512 + n0 + ml] = acc[r];
}

// --------------------------------------------------------------------------
// K5: per-batch s = sigmoid(X[b] @ f2c[b] + bmat); l[b] = v @ s   (24x24)
// --------------------------------------------------------------------------
__global__ __launch_bounds__(576) void k5_attn(
    const float* __restrict__ X, const float* __restrict__ f2c,
    const float* __restrict__ bmat, const float* __restrict__ vmat,
    float* __restrict__ lbuf)
{
  __shared__ float F2s[C * T];              // 48 KB
  __shared__ float ss[T * T];
  const int b = blockIdx.x;
  const int tid = threadIdx.x;
  for (int i = tid; i < C * T; i += 576) F2s[i] = f2c[b * C * T + i];
  __syncthreads();
  const int i = tid / T, j = tid % T;
  const float* xr = X + (b * T + i) * C;
  float a = 0.f;
  for (int c = 0; c < C; ++c) a += xr[c] * F2s[c * T + j];
  ss[i * T + j] = 1.f / (1.f + expf(-(a + bmat[i * T + j])));
  __syncthreads();
  float a2 = 0.f;
#pragma unroll
  for (int j2 = 0; j2 < T; ++j2) a2 += vmat[i * T + j2] * ss[j2 * T + j];
  lbuf[(b * T + i) * T + j] = a2;
}

// --------------------------------------------------------------------------
// K6: BatchNorm (batch stats per channel k over 192 samples) + block mask +
// row softmax. Single block; output is only 4608 floats.
// --------------------------------------------------------------------------
__global__ __launch_bounds__(576) void k6_bn_softmax(
    const float* __restrict__ lbuf, const float* __restrict__ gamma,
    const float* __restrict__ beta, float* __restrict__ out)
{
  __shared__ float L[B * T * T];            // 4608
  __shared__ float sc[T], sh[T];
  const int tid = threadIdx.x;
  for (int i = tid; i < B * T * T; i += 576) L[i] = lbuf[i];
  __syncthreads();
  if (tid < T) {
    const int k = tid;
    float s = 0.f, s2 = 0.f;
    for (int m = 0; m < B * T; ++m) { const float v = L[m * T + k]; s += v; s2 += v * v; }
    const float mean = s * (1.f / (B * T));
    const float var  = s2 * (1.f / (B * T)) - mean * mean;
    const float scale = gamma[k] * rsqrtf(var + 1e-5f);
    sc[k] = scale;
    sh[k] = beta[k] - mean * scale;
  }
  __syncthreads();
  for (int idx = tid; idx < B * T * T; idx += 576) {
    const int k = idx % T;
    const int i = (idx / T) % T;
    float v = L[idx] * sc[k] + sh[k];
    if ((i < 12) != (k < 12)) v -= 1e13f;   // block-diagonal additive mask
    L[idx] = v;
  }
  __syncthreads();
  if (tid < B * T) {
    const int base = tid * T;
    float mx = -1e30f;
    for (int k = 0; k < T; ++k) mx = fmaxf(mx, L[base + k]);
    float s = 0.f;
    for (int k = 0; k < T; ++k) s += expf(L[base + k] - mx);
    const float inv = 1.f / s;
    for (int k = 0; k < T; ++k) out[base + k] = expf(L[base + k] - mx) * inv;
  }
}

// --------------------------------------------------------------------------
extern "C" void kernel_launch(void* const* d_in, const int* in_sizes, int n_in,
                              void* d_out, int out_size, void* d_ws, size_t ws_size,
                              hipStream_t stream)
{
  const float* seq  = (const float*)d_in[0];
  const float* w1   = (const float*)d_in[1];
  const float* w2   = (const float*)d_in[2];
  const float* Wd1  = (const float*)d_in[3];
  const float* Wd2  = (const float*)d_in[4];
  const float* wmat = (const float*)d_in[5];
  const float* bmat = (const float*)d_in[6];
  const float* vmat = (const float*)d_in[7];
  const float* gam  = (const float*)d_in[8];
  const float* bet  = (const float*)d_in[9];
  float* out = (float*)d_out;

  // workspace carve-out (~15 MB of fp32)
  float* ws     = (float*)d_ws;
  float* f1part = ws;                             // CS*B*T*N    = 1,572,864
  float* f1c    = f1part + (size_t)CS * B * T * N; // B*T*N      =   393,216
  float* f2part = f1c    + (size_t)B * T * N;      // B*NCH*C*T  = 1,572,864
  float* f2r    = f2part + (size_t)B * NCH * C * T;// B*C*T      =    98,304
  float* f2c    = f2r    + (size_t)B * C * T;      // B*C*T      =    98,304
  float* X      = f2c    + (size_t)B * C * T;      // B*T*C      =    98,304
  float* lbuf   = X      + (size_t)B * T * C;      // B*T*T      =     4,608

  k1_stream  <<<CS * B * NCH, 256, 0, stream>>>(seq, w1, w2, f1part, f2part);
  k2a_reduce <<<(B * C * T) / 256, 256, 0, stream>>>(f2part, f2r);
  k2b_convd2 <<<B * 8, 256, 0, stream>>>(f2r, Wd2, f2c);
  k3_convd1  <<<B * NCH, 256, 0, stream>>>(f1part, Wd1, f1c);
  k4_wmma_gemm<<<12 * 32, 32, 0, stream>>>(f1c, wmat, X);
  k5_attn    <<<B, 576, 0, stream>>>(X, f2c, bmat, vmat, lbuf);
  k6_bn_softmax<<<1, 576, 0, stream>>>(lbuf, gam, bet, out);
}